// BatchSampler_77704548319374
// MI455X (gfx1250) — compile-verified
//
#include <hip/hip_runtime.h>
#include <stdint.h>

#define NT     1024          // 32 waves of 32 (wave32)
#define CHUNK  8192          // floats per streamed LDS chunk (32 KB)
#define NB1    4096          // level-1/2 radix bins (12 bits)
#define NB2    4096
#define NB3    256           // level-3 bins (8 bits)

#if defined(__HIP_DEVICE_COMPILE__)
# if __has_builtin(__builtin_amdgcn_tensor_load_to_lds) && __has_builtin(__builtin_amdgcn_s_wait_tensorcnt)
#  define HAS_TDM 1
# endif
#endif
#ifndef HAS_TDM
# define HAS_TDM 0
#endif

typedef unsigned int v4u __attribute__((ext_vector_type(4)));
typedef int          v8i __attribute__((ext_vector_type(8)));
typedef int          v4i __attribute__((ext_vector_type(4)));

// Monotone float -> uint key: x >= y  <=>  ordkey(x) >= ordkey(y)
__device__ __forceinline__ unsigned int ordkey(float x) {
  unsigned int b = __float_as_uint(x);
  unsigned int m = (unsigned int)(((int)b) >> 31) | 0x80000000u;
  return b ^ m;
}

__device__ __forceinline__ unsigned int ldsAddrOf(const void* p) {
  // low 32 bits of a flat pointer into LDS = byte offset within the group's LDS
  return (unsigned int)(uintptr_t)p;
}

#if HAS_TDM
// Issue one TDM load of CHUNK f32 elements (OOB beyond `validElems` returns 0)
__device__ __forceinline__ void tdm_issue(float* ldsDst, const float* gsrc, unsigned int validElems) {
  unsigned long long ga = (unsigned long long)(uintptr_t)gsrc;
  v4u g0;
  g0.x = 1u;                                                    // count=1, user D#
  g0.y = ldsAddrOf(ldsDst);                                     // lds_addr
  g0.z = (unsigned int)ga;                                      // global_addr[31:0]
  g0.w = (unsigned int)((ga >> 32) & 0x01FFFFFFu) | (2u << 30); // addr[56:32] | type=2
  unsigned int d0 = validElems;
  v8i g1;
  g1[0] = (int)(2u << 16);                               // data_size = 4B
  g1[1] = (int)((d0 & 0xFFFFu) << 16);                   // tensor_dim0 lo16 (bits 63:48)
  g1[2] = (int)(((d0 >> 16) & 0xFFFFu) | (1u << 16));    // dim0 hi16 | tensor_dim1=1 lo16
  g1[3] = (int)(((unsigned int)CHUNK) << 16);            // dim1 hi16=0 | tile_dim0=CHUNK
  g1[4] = 0;                                             // tile_dim1=0 (1D), tile_dim2=0
  g1[5] = (int)d0;                                       // tensor_dim0_stride lo32
  g1[6] = 0;                                             // stride hi16 | dim1_stride lo16
  g1[7] = 0;
  v4i z4 = 0;
#if __clang_major__ >= 23
  v8i z8 = 0;
  __builtin_amdgcn_tensor_load_to_lds(g0, g1, z4, z4, z8, 0);
#else
  __builtin_amdgcn_tensor_load_to_lds(g0, g1, z4, z4, 0);
#endif
}
#endif

// Stream a row through LDS in CHUNK-sized double-buffered pieces (TDM DMA).
template <typename Fn>
__device__ void streamRow(const float* row, int V, float* lbuf, Fn fn) {
  const int nC = (V + CHUNK - 1) / CHUNK;
#if HAS_TDM
  const bool issuer = (threadIdx.x < 32);   // wave 0 drives the TDM
  if (issuer) tdm_issue(lbuf, row, (unsigned int)V);
  for (int c = 0; c < nC; ++c) {
    float* buf = lbuf + (size_t)(c & 1) * CHUNK;
    if (issuer) {
      if (c + 1 < nC) {
        tdm_issue(lbuf + (size_t)((c + 1) & 1) * CHUNK,
                  row + (size_t)(c + 1) * CHUNK,
                  (unsigned int)(V - (c + 1) * CHUNK));
        __builtin_amdgcn_s_wait_tensorcnt(1);   // chunk c complete (in-order)
      } else {
        __builtin_amdgcn_s_wait_tensorcnt(0);
      }
    }
    __syncthreads();
    const int base = c * CHUNK;
    const int cnt = (V - base < CHUNK) ? (V - base) : CHUNK;
    fn(buf, base, cnt);
    __syncthreads();
  }
#else
  for (int c = 0; c < nC; ++c) {
    float* buf = lbuf + (size_t)(c & 1) * CHUNK;
    const int base = c * CHUNK;
    const int cnt = (V - base < CHUNK) ? (V - base) : CHUNK;
    for (int i = threadIdx.x; i < cnt; i += NT) buf[i] = row[base + i];
    __syncthreads();
    fn(buf, base, cnt);
    __syncthreads();
  }
#endif
}

// Vectorized per-element map over one LDS chunk: g(globalIdx, scaledValue).
// ds_load_b128 for the bulk, scalar tail.
template <typename G>
__device__ __forceinline__ void forEach4(const float* buf, int base, int cnt, float invT, G g) {
  const float4* b4 = (const float4*)buf;
  const int n4 = cnt >> 2;
  for (int q = threadIdx.x; q < n4; q += NT) {
    float4 v = b4[q];
    int i = base + (q << 2);
    g(i + 0, v.x * invT);
    g(i + 1, v.y * invT);
    g(i + 2, v.z * invT);
    g(i + 3, v.w * invT);
  }
  for (int i = (n4 << 2) + (int)threadIdx.x; i < cnt; i += NT) g(base + i, buf[i] * invT);
}

__device__ __forceinline__ void zeroHist(unsigned long long* h, int n) {
  for (int i = threadIdx.x; i < n; i += NT) h[i] = 0ull;
}

__device__ __forceinline__ float blockMax(float v, float* sw) {
  for (int off = 16; off > 0; off >>= 1) v = fmaxf(v, __shfl_down(v, off, 32));
  const int wid = threadIdx.x >> 5, lane = threadIdx.x & 31;
  if (lane == 0) sw[wid] = v;
  __syncthreads();
  if (wid == 0) {
    v = sw[lane];
    for (int off = 16; off > 0; off >>= 1) v = fmaxf(v, __shfl_down(v, off, 32));
    if (lane == 0) sw[0] = v;
  }
  __syncthreads();
  float r = sw[0];
  __syncthreads();
  return r;
}

struct SelRes { int bin; unsigned long long above, total, target; };

// Find the LARGEST bin b such that suffix-sum S(b)=sum(hist[b..NB-1]) satisfies
// (strict ? S>target : S>=target). Returns the bin, S(b+1) ("above"), total, target.
// Wave32 shfl scans: only 5 block barriers total.
__device__ SelRes histSelect(unsigned long long* hist, int NB,
                             unsigned long long* swv,   // [32] u64 scratch
                             int* s_ibin, unsigned long long* s_above,
                             unsigned long long absTarget, double frac, bool useFrac,
                             bool strict) {
  const int tid = threadIdx.x;
  const int lane = tid & 31, wid = tid >> 5;
  const int per = (NB + NT - 1) / NT;
  const int d0 = tid * per;             // d: descending-bin index, d=0 is bin NB-1
  unsigned long long lsum = 0ull;
  for (int j = 0; j < per; ++j) {
    int d = d0 + j;
    if (d < NB) lsum += hist[NB - 1 - d];
  }
  // intra-wave inclusive scan (no barriers)
  unsigned long long inc = lsum;
  for (int off = 1; off < 32; off <<= 1) {
    unsigned long long u = __shfl_up(inc, off, 32);
    if (lane >= off) inc += u;
  }
  if (lane == 31) swv[wid] = inc;
  __syncthreads();
  if (wid == 0) {                       // cross-wave scan of 32 wave totals
    unsigned long long t = swv[lane];
    for (int off = 1; off < 32; off <<= 1) {
      unsigned long long u = __shfl_up(t, off, 32);
      if (lane >= off) t += u;
    }
    swv[lane] = t;
  }
  if (tid == 0) { *s_ibin = 0x7FFFFFFF; *s_above = 0ull; }
  __syncthreads();
  const unsigned long long waveExcl = (wid > 0) ? swv[wid - 1] : 0ull;
  const unsigned long long total = swv[31];
  const unsigned long long excl = waveExcl + (inc - lsum);
  const unsigned long long target =
      useFrac ? (unsigned long long)((double)total * frac) : absTarget;
  unsigned long long run = excl;
  int fd = -1; unsigned long long fab = 0ull;
  for (int j = 0; j < per; ++j) {
    int d = d0 + j;
    if (d >= NB) break;
    unsigned long long s = run + hist[NB - 1 - d];
    bool c = strict ? (s > target) : (s >= target);
    if (c) { fd = d; fab = run; break; }
    run = s;
  }
  if (fd >= 0) atomicMin(s_ibin, fd);
  __syncthreads();
  const int dsel = *s_ibin;
  if (fd == dsel && fd >= 0) *s_above = fab;
  __syncthreads();
  SelRes r;
  r.bin = (dsel == 0x7FFFFFFF) ? 0 : (NB - 1 - dsel);
  r.above = *s_above;
  r.total = total;
  r.target = target;
  __syncthreads();
  return r;
}

// JAX threefry2x32 (ks2 = k0^k1^0x1BD11BDA), 20 rounds
__device__ __forceinline__ unsigned int rotl32(unsigned int v, int r) {
  return (v << r) | (v >> (32 - r));
}
__device__ __forceinline__ void threefry2x32(unsigned int k0, unsigned int k1,
                                             unsigned int x0, unsigned int x1,
                                             unsigned int& o0, unsigned int& o1) {
  unsigned int ks[3] = {k0, k1, k0 ^ k1 ^ 0x1BD11BDAu};
  x0 += ks[0]; x1 += ks[1];
  const int rotA[4] = {13, 15, 26, 6};
  const int rotB[4] = {17, 29, 16, 24};
#pragma unroll
  for (int i = 0; i < 5; ++i) {
    const int* rr = (i & 1) ? rotB : rotA;
#pragma unroll
    for (int j = 0; j < 4; ++j) { x0 += x1; x1 = rotl32(x1, rr[j]); x1 ^= x0; }
    x0 += ks[(i + 1) % 3];
    x1 += ks[(i + 2) % 3] + (unsigned int)(i + 1);
  }
  o0 = x0; o1 = x1;
}

// Gumbel noise matching jax.random.categorical(key(42), ...) counter layout.
__device__ __forceinline__ float gumbel42(unsigned long long g, unsigned long long n2) {
  unsigned int c0, c1, o0, o1;
  int sel;
  if (g < n2) { c0 = (unsigned int)g; c1 = (unsigned int)(g + n2); sel = 0; }
  else        { c0 = (unsigned int)(g - n2); c1 = (unsigned int)g; sel = 1; }
  threefry2x32(0u, 42u, c0, c1, o0, o1);
  unsigned int bits = sel ? o1 : o0;
  float u01 = __uint_as_float((bits >> 9) | 0x3F800000u) - 1.0f;
  const float tiny = 1.1754944e-38f;
  float u = fmaxf(tiny, u01 * (1.0f - tiny) + tiny);
  return -__logf(-__logf(u));
}

__global__ void __launch_bounds__(NT)
BatchSampler_77704548319374_kernel(const float* __restrict__ logits,
                                   const float* __restrict__ temps,
                                   const float* __restrict__ topps,
                                   const long long* __restrict__ topks,
                                   const float* __restrict__ minps,
                                   float* __restrict__ out,
                                   int V, unsigned long long n2) {
  __shared__ __align__(16) float lbuf[2 * CHUNK];   // 64 KB streaming buffers
  __shared__ unsigned long long hist[NB1];          // 32 KB radix histogram
  __shared__ unsigned long long swv[32];            // cross-wave scan scratch
  __shared__ int s_ibin;
  __shared__ unsigned long long s_above;
  __shared__ float s_f[33];
  __shared__ float s_wS[32];
  __shared__ int s_wI[32];

  const int row = blockIdx.x;
  const int tid = threadIdx.x;
  const float invT = 1.0f / fmaxf(temps[row], 1e-8f);
  const float topp = topps[row];
  const float minp = minps[row];
  const long long kll = topks[row];
  const bool activeK = (kll > 0 && kll < (long long)V);
  const unsigned long long k1t = (unsigned long long)(kll > 0 ? kll : 1);
  const float* rowp = logits + (size_t)row * (size_t)V;

  // ---------- pass 1: row max + level-1 count histogram ----------
  zeroHist(hist, NB1);
  __syncthreads();
  float lmax = -3.402823466e38f;
  streamRow(rowp, V, lbuf, [&](const float* buf, int base, int cnt) {
    forEach4(buf, base, cnt, invT, [&](int idx, float x) {
      (void)idx;
      lmax = fmaxf(lmax, x);
      atomicAdd(&hist[ordkey(x) >> 20], 1ull);
    });
  });
  const float M = blockMax(lmax, s_f);

  // ---------- top-k threshold via 3-level radix select ----------
  unsigned int tkKey = 0u;
  if (activeK) {
    SelRes s1 = histSelect(hist, NB1, swv, &s_ibin, &s_above, k1t, 0.0, false, false);
    unsigned long long k2 = k1t - s1.above;
    unsigned int b1 = (unsigned int)s1.bin;
    zeroHist(hist, NB2);
    __syncthreads();
    streamRow(rowp, V, lbuf, [&](const float* buf, int base, int cnt) {
      forEach4(buf, base, cnt, invT, [&](int idx, float x) {
        (void)idx;
        unsigned int kk = ordkey(x);
        if ((kk >> 20) == b1) atomicAdd(&hist[(kk >> 8) & 0xFFFu], 1ull);
      });
    });
    SelRes s2 = histSelect(hist, NB2, swv, &s_ibin, &s_above, k2, 0.0, false, false);
    unsigned long long k3 = k2 - s2.above;
    unsigned int pre24 = (b1 << 12) | (unsigned int)s2.bin;
    zeroHist(hist, NB3);
    __syncthreads();
    streamRow(rowp, V, lbuf, [&](const float* buf, int base, int cnt) {
      forEach4(buf, base, cnt, invT, [&](int idx, float x) {
        (void)idx;
        unsigned int kk = ordkey(x);
        if ((kk >> 8) == pre24) atomicAdd(&hist[kk & 0xFFu], 1ull);
      });
    });
    SelRes s3 = histSelect(hist, NB3, swv, &s_ibin, &s_above, k3, 0.0, false, false);
    tkKey = (pre24 << 8) | (unsigned int)s3.bin;
  }

  // ---------- top-p threshold: mass-weighted radix select ----------
  const float MASS_SCALE = 1.7592186044416e13f;  // 2^44 fixed point (deterministic u64)
  zeroHist(hist, NB1);
  __syncthreads();
  streamRow(rowp, V, lbuf, [&](const float* buf, int base, int cnt) {
    forEach4(buf, base, cnt, invT, [&](int idx, float x) {
      (void)idx;
      unsigned int kk = ordkey(x);
      if (kk >= tkKey) {
        unsigned long long w = (unsigned long long)(__expf(x - M) * MASS_SCALE);
        atomicAdd(&hist[kk >> 20], w);
      }
    });
  });
  SelRes m1 = histSelect(hist, NB1, swv, &s_ibin, &s_above, 0ull, (double)topp, true, true);
  unsigned long long t2 = m1.target - m1.above;
  unsigned int mb1 = (unsigned int)m1.bin;
  zeroHist(hist, NB2);
  __syncthreads();
  streamRow(rowp, V, lbuf, [&](const float* buf, int base, int cnt) {
    forEach4(buf, base, cnt, invT, [&](int idx, float x) {
      (void)idx;
      unsigned int kk = ordkey(x);
      if (kk >= tkKey && (kk >> 20) == mb1) {
        unsigned long long w = (unsigned long long)(__expf(x - M) * MASS_SCALE);
        atomicAdd(&hist[(kk >> 8) & 0xFFFu], w);
      }
    });
  });
  SelRes m2 = histSelect(hist, NB2, swv, &s_ibin, &s_above, t2, 0.0, false, true);
  unsigned long long t3 = t2 - m2.above;
  unsigned int mpre24 = (mb1 << 12) | (unsigned int)m2.bin;
  zeroHist(hist, NB3);
  __syncthreads();
  streamRow(rowp, V, lbuf, [&](const float* buf, int base, int cnt) {
    forEach4(buf, base, cnt, invT, [&](int idx, float x) {
      (void)idx;
      unsigned int kk = ordkey(x);
      if (kk >= tkKey && (kk >> 8) == mpre24) {
        unsigned long long w = (unsigned long long)(__expf(x - M) * MASS_SCALE);
        atomicAdd(&hist[kk & 0xFFu], w);
      }
    });
  });
  SelRes m3 = histSelect(hist, NB3, swv, &s_ibin, &s_above, t3, 0.0, false, true);
  unsigned int tpKey = (mpre24 << 8) | (unsigned int)m3.bin;

  // ---------- min-p threshold (pure logit threshold; Z cancels) ----------
  unsigned int tmKey = 0u;
  if (minp > 0.0f) tmKey = ordkey(M + __logf(minp));

  unsigned int tf = tkKey;
  if (tpKey > tf) tf = tpKey;
  if (tmKey > tf) tf = tmKey;

  // ---------- Gumbel-argmax sample over kept tokens ----------
  float bs = -3.402823466e38f;
  int bi = 0x7FFFFFFF;
  streamRow(rowp, V, lbuf, [&](const float* buf, int base, int cnt) {
    forEach4(buf, base, cnt, invT, [&](int idx, float x) {
      if (ordkey(x) >= tf) {
        unsigned long long g =
            (unsigned long long)row * (unsigned long long)V + (unsigned long long)idx;
        float sc = x + gumbel42(g, n2);
        if (sc > bs || (sc == bs && idx < bi)) { bs = sc; bi = idx; }
      }
    });
  });
  // block argmax (tie -> lowest index), wave32 shuffles
  for (int off = 16; off > 0; off >>= 1) {
    float os = __shfl_down(bs, off, 32);
    int oi = __shfl_down(bi, off, 32);
    if (os > bs || (os == bs && oi < bi)) { bs = os; bi = oi; }
  }
  const int wid = tid >> 5, lane = tid & 31;
  if (lane == 0) { s_wS[wid] = bs; s_wI[wid] = bi; }
  __syncthreads();
  if (wid == 0) {
    bs = s_wS[lane];
    bi = s_wI[lane];
    for (int off = 16; off > 0; off >>= 1) {
      float os = __shfl_down(bs, off, 32);
      int oi = __shfl_down(bi, off, 32);
      if (os > bs || (os == bs && oi < bi)) { bs = os; bi = oi; }
    }
    if (lane == 0) out[row] = (float)bi;
  }
}

extern "C" void kernel_launch(void* const* d_in, const int* in_sizes, int n_in,
                              void* d_out, int out_size, void* d_ws, size_t ws_size,
                              hipStream_t stream) {
  (void)n_in; (void)out_size; (void)d_ws; (void)ws_size;
  const float* logits = (const float*)d_in[0];
  const float* temps = (const float*)d_in[1];
  const float* topps = (const float*)d_in[2];
  const long long* topks = (const long long*)d_in[3];
  const float* minps = (const float*)d_in[4];
  float* out = (float*)d_out;
  const int B = in_sizes[1];
  const int V = in_sizes[0] / B;
  const unsigned long long n2 = ((unsigned long long)B * (unsigned long long)V) / 2ull;
  hipLaunchKernelGGL(BatchSampler_77704548319374_kernel, dim3(B), dim3(NT), 0, stream,
                     logits, temps, topps, topks, minps, out, V, n2);
}